// SVDO_35107062678008
// MI455X (gfx1250) — compile-verified
//
#include <hip/hip_runtime.h>
#include <hip/hip_bf16.h>
#include <math.h>

// ---------------------------------------------------------------------------
// R = proj_SO3(x @ W^T + b) per 3x3 matrix.
// GEMM: V_WMMA_F32_16X16X4_F32; x streamed global->LDS with
//       GLOBAL_LOAD_ASYNC_TO_LDS_B128 (ASYNCcnt), double-buffered so the
//       copy of chunk c+1 overlaps the WMMAs of chunk c.
// SVD:  M^T M = V diag(l) V^T via unrolled 3x3 Jacobi; R = M V diag(d) V^T,
//       d = rsqrt(l) with sign(det M) applied to the smallest eigenvalue.
// ---------------------------------------------------------------------------

typedef __attribute__((ext_vector_type(2))) float v2f;
typedef __attribute__((ext_vector_type(8))) float v8f;

#define TILE_ROWS 128   // batch rows per block
#define KCHUNK    16    // K elements staged per chunk
#define KDIM      512
#define NCH       (KDIM / KCHUNK)   // 32 chunks
#define XPAD      20    // padded row stride in floats (16B aligned, conflict-free)

__device__ __forceinline__ void jacobi_rot(float A[3][3], float V[3][3],
                                           const int p, const int q) {
    float apq = A[p][q];
    float c = 1.0f, s = 0.0f;
    if (fabsf(apq) > 1e-28f) {
        float theta = 0.5f * (A[q][q] - A[p][p]) / apq;
        float t = copysignf(1.0f, theta) /
                  (fabsf(theta) + sqrtf(1.0f + theta * theta));
        c = 1.0f / sqrtf(1.0f + t * t);
        s = t * c;
    }
#pragma unroll
    for (int k = 0; k < 3; ++k) {         // A := A * J
        float akp = A[k][p], akq = A[k][q];
        A[k][p] = c * akp - s * akq;
        A[k][q] = s * akp + c * akq;
    }
#pragma unroll
    for (int k = 0; k < 3; ++k) {         // A := J^T * A
        float apk = A[p][k], aqk = A[q][k];
        A[p][k] = c * apk - s * aqk;
        A[q][k] = s * apk + c * aqk;
    }
#pragma unroll
    for (int k = 0; k < 3; ++k) {         // V := V * J
        float vkp = V[k][p], vkq = V[k][q];
        V[k][p] = c * vkp - s * vkq;
        V[k][q] = s * vkp + c * vkq;
    }
}

__global__ __launch_bounds__(256)
void svdo_rot_kernel(const float* __restrict__ x,
                     const float* __restrict__ W,
                     const float* __restrict__ bvec,
                     float* __restrict__ out) {
    __shared__ float  xtile[2][TILE_ROWS * XPAD];     // 2 x 10,240 B
    __shared__ float2 wt2[(KDIM / 2) * 16];           // 32,768 B  W^T as K-pairs
    __shared__ float  outm[TILE_ROWS * 9];            //  4,608 B

    const int tid  = threadIdx.x;
    const int lane = tid & 31;
    const int wave = tid >> 5;            // 0..7
    const int hi   = lane >> 4;           // lane group (K half select)
    const int lo   = lane & 15;           // M for A-frag, N for B/C-frag
    const int rowBlock = blockIdx.x * TILE_ROWS;

    // ---- stage padded W^T into LDS: wt2[(K/2)*16 + n] = {W[n][K], W[n][K+1]}
#pragma unroll
    for (int i = 0; i < 16; ++i) {
        int p   = tid + i * 256;          // 0..4095
        int kk2 = p >> 4;                 // K pair index (K = 2*kk2)
        int nn  = p & 15;
        float2 v = make_float2(0.0f, 0.0f);
        if (nn < 9)
            v = *reinterpret_cast<const float2*>(&W[(size_t)nn * KDIM + 2 * kk2]);
        wt2[p] = v;
    }

    // ---- async-copy helper: 128 x 16 f32 chunk, 2 x B128 per thread
    auto stage_chunk = [&](int ch, int buf) {
#pragma unroll
        for (int i = 0; i < 2; ++i) {
            int q   = tid + (i << 8);     // float4 index 0..511
            int row = q >> 2;             // 4 float4 per row
            int c4  = q & 3;
            const float* g =
                &x[(size_t)(rowBlock + row) * KDIM + ch * KCHUNK + c4 * 4];
            // LDS byte offset = low 32 bits of flat shared pointer
            uint32_t loff = (uint32_t)(uintptr_t)&xtile[buf][row * XPAD + c4 * 4];
            uint64_t gaddr = (uint64_t)(uintptr_t)g;
            asm volatile("global_load_async_to_lds_b128 %0, %1, off"
                         :: "v"(loff), "v"(gaddr)
                         : "memory");
        }
    };

    stage_chunk(0, 0);                    // prologue: chunk 0 in flight

    v8f acc = {};                          // 16x16 f32 accumulator (cols 0..8 valid)
    const int mrow = (wave << 4) + lo;     // this lane's A-matrix row in block tile

    for (int ch = 0; ch < NCH; ++ch) {
        const int cur = ch & 1;
        if (ch + 1 < NCH) {
            stage_chunk(ch + 1, cur ^ 1);  // keep next chunk in flight
            asm volatile("s_wait_asynccnt 2" ::: "memory");  // chunk ch done
        } else {
            asm volatile("s_wait_asynccnt 0" ::: "memory");
        }
        __syncthreads();                   // publish chunk ch to all waves

        // ---- 4 x V_WMMA_F32_16X16X4_F32 over this chunk
#pragma unroll
        for (int j = 0; j < 4; ++j) {
            int kl = 4 * j;                                // local K base
            int kg = ch * KCHUNK + kl;                     // global K base
            // A frag: lane lo = row M, VGPR0/1 = K = kg+2*hi, kg+2*hi+1
            float2 av = *reinterpret_cast<const float2*>(
                &xtile[cur][mrow * XPAD + kl + 2 * hi]);
            // B frag: lane lo = col N, K pair at kg/2 + hi
            float2 bv = wt2[((kg >> 1) + hi) * 16 + lo];
            v2f a; a.x = av.x; a.y = av.y;
            v2f b; b.x = bv.x; b.y = bv.y;
            acc = __builtin_amdgcn_wmma_f32_16x16x4_f32(
                false, a, false, b, (short)0, acc, false, false);
        }
        __syncthreads();                   // protect buffer before next overwrite
    }

    // ---- scatter C (+bias) to LDS: lane holds col N=lo, rows r+8*hi
    if (lo < 9) {
        float bias = bvec[lo];
#pragma unroll
        for (int r = 0; r < 8; ++r) {
            int m = (wave << 4) + r + (hi << 3);
            outm[m * 9 + lo] = acc[r] + bias;
        }
    }
    __syncthreads();

    // ---- per-matrix nearest-rotation via 3x3 symmetric Jacobi
    if (tid < TILE_ROWS) {
        float M[3][3];
#pragma unroll
        for (int i = 0; i < 3; ++i)
#pragma unroll
            for (int j = 0; j < 3; ++j)
                M[i][j] = outm[tid * 9 + i * 3 + j];

        float det = M[0][0] * (M[1][1] * M[2][2] - M[1][2] * M[2][1])
                  - M[0][1] * (M[1][0] * M[2][2] - M[1][2] * M[2][0])
                  + M[0][2] * (M[1][0] * M[2][1] - M[1][1] * M[2][0]);

        float A[3][3];
#pragma unroll
        for (int i = 0; i < 3; ++i)
#pragma unroll
            for (int j = 0; j < 3; ++j)
                A[i][j] = M[0][i] * M[0][j] + M[1][i] * M[1][j] + M[2][i] * M[2][j];

        float V[3][3] = {{1.f, 0.f, 0.f}, {0.f, 1.f, 0.f}, {0.f, 0.f, 1.f}};
#pragma unroll
        for (int sweep = 0; sweep < 4; ++sweep) {
            jacobi_rot(A, V, 0, 1);
            jacobi_rot(A, V, 0, 2);
            jacobi_rot(A, V, 1, 2);
        }

        float l0 = fmaxf(A[0][0], 1e-30f);
        float l1 = fmaxf(A[1][1], 1e-30f);
        float l2 = fmaxf(A[2][2], 1e-30f);
        float d0 = rsqrtf(l0), d1 = rsqrtf(l1), d2 = rsqrtf(l2);
        float sgn = (det < 0.0f) ? -1.0f : 1.0f;
        if (l0 <= l1 && l0 <= l2)      d0 *= sgn;   // sign on smallest sigma
        else if (l1 <= l2)             d1 *= sgn;
        else                           d2 *= sgn;

        float S[3][3];                  // V diag(d) V^T
#pragma unroll
        for (int i = 0; i < 3; ++i)
#pragma unroll
            for (int j = 0; j < 3; ++j)
                S[i][j] = V[i][0] * d0 * V[j][0]
                        + V[i][1] * d1 * V[j][1]
                        + V[i][2] * d2 * V[j][2];

        size_t orow = (size_t)(rowBlock + tid) * 9;
#pragma unroll
        for (int i = 0; i < 3; ++i)
#pragma unroll
            for (int j = 0; j < 3; ++j)
                out[orow + i * 3 + j] =
                    M[i][0] * S[0][j] + M[i][1] * S[1][j] + M[i][2] * S[2][j];
    }
}

extern "C" void kernel_launch(void* const* d_in, const int* in_sizes, int n_in,
                              void* d_out, int out_size, void* d_ws, size_t ws_size,
                              hipStream_t stream) {
    const float* x = (const float*)d_in[0];
    const float* W = (const float*)d_in[1];
    const float* b = (const float*)d_in[2];
    float* out = (float*)d_out;

    int rows   = in_sizes[0] / KDIM;          // 131072
    int blocks = rows / TILE_ROWS;            // 1024

    svdo_rot_kernel<<<blocks, 256, 0, stream>>>(x, W, b, out);
}